// HardNegativeContrastiveLoss_6846177870109
// MI455X (gfx1250) — compile-verified
//
#include <hip/hip_runtime.h>

typedef float v2f __attribute__((ext_vector_type(2)));
typedef float v8f __attribute__((ext_vector_type(8)));
typedef unsigned int u32x4 __attribute__((ext_vector_type(4)));
typedef int i32x4 __attribute__((ext_vector_type(4)));
typedef int i32x8 __attribute__((ext_vector_type(8)));

#define BATCH 8192
#define DIM   1024
#define INV_T (1.0f / 0.07f)
#define TOPM  0.95f
#define BOTM  0.05f

#define TM  32              // rows per block
#define TN  256             // columns per tile iteration (32 per wave)
#define KB  32              // K-block staged in LDS
#define LDK (KB + 2)        // stride 34: matches TDM pad (32 DWORDs + 2 DWORDs)
#define NK  (DIM / KB)      // 32 K-blocks
#define NEG_INF (-__builtin_inff())

// ---------------------------------------------------------------------------
// TDM: issue a 2-D tile DMA global -> LDS.  rows x 32 floats, row stride DIM.
// D# per CDNA5 ISA §8.3/8.4.  Padding: +2 DWORDs every 32 DWORDs -> LDK=34.
// ---------------------------------------------------------------------------
__device__ __forceinline__ void tdm_load_tile(const float* gptr, unsigned ldsOff,
                                              int rows) {
  const unsigned long long ga = (unsigned long long)(uintptr_t)gptr;
  u32x4 g0;
  g0[0] = 1u;                                    // count=1 (valid), flags 0
  g0[1] = ldsOff;                                // lds_addr (bytes)
  g0[2] = (unsigned)(ga & 0xFFFFFFFFu);          // global_addr[31:0]
  g0[3] = (unsigned)((ga >> 32) & 0x01FFFFFFu)   // global_addr[56:32]
          | (2u << 30);                          // type = 2 ("image")
  i32x8 g1;
  g1[0] = (2 << 16)      // data_size = 4 bytes
        | (1 << 20)      // pad_enable
        | (4 << 22)      // pad_interval: code 4 -> every 32 DWORDs (one row)
        | (1 << 25);     // pad_amount:  code 1 -> 2 DWORDs
  g1[1] = (int)((unsigned)(DIM & 0xFFFF) << 16);   // tensor_dim0[15:0]
  g1[2] = (int)((unsigned)(DIM >> 16)              // tensor_dim0[31:16]
        | ((unsigned)(BATCH & 0xFFFF) << 16));     // tensor_dim1[15:0]
  g1[3] = (int)((unsigned)(BATCH >> 16)            // tensor_dim1[31:16]
        | (32u << 16));                            // tile_dim0 = 32 elements
  g1[4] = rows;                                    // tile_dim1 = rows, tile_dim2=0
  g1[5] = DIM;                                     // tensor_dim0_stride[31:0]
  g1[6] = 0;                                       // stride0 hi, stride1 lo
  g1[7] = 0;                                       // stride1 hi
  const i32x4 z4 = {0, 0, 0, 0};                   // 2-D: groups 2/3 unused
  const i32x8 z8 = {0, 0, 0, 0, 0, 0, 0, 0};       // extra group (clang-23 form)
  __builtin_amdgcn_tensor_load_to_lds(g0, g1, z4, z4, z8, 0);
}

// ---------------------------------------------------------------------------
// Kernel 1: per-row inverse norms + positive similarity pos[i]
// ---------------------------------------------------------------------------
__global__ __launch_bounds__(256) void norms_kernel(const float* __restrict__ A,
                                                    const float* __restrict__ B,
                                                    float* __restrict__ invA,
                                                    float* __restrict__ invB,
                                                    float* __restrict__ pos) {
  const int row = blockIdx.x;
  const int t   = threadIdx.x;
  const float4 av = ((const float4*)(A + (size_t)row * DIM))[t];
  const float4 bv = ((const float4*)(B + (size_t)row * DIM))[t];
  float aa = av.x * av.x + av.y * av.y + av.z * av.z + av.w * av.w;
  float bb = bv.x * bv.x + bv.y * bv.y + bv.z * bv.z + bv.w * bv.w;
  float ab = av.x * bv.x + av.y * bv.y + av.z * bv.z + av.w * bv.w;
#pragma unroll
  for (int off = 16; off > 0; off >>= 1) {
    aa += __shfl_xor(aa, off);
    bb += __shfl_xor(bb, off);
    ab += __shfl_xor(ab, off);
  }
  __shared__ float sA[8], sB[8], sC[8];
  const int w = t >> 5, lane = t & 31;
  if (lane == 0) { sA[w] = aa; sB[w] = bb; sC[w] = ab; }
  __syncthreads();
  if (t == 0) {
    float a = 0.f, b = 0.f, c = 0.f;
#pragma unroll
    for (int j = 0; j < 8; ++j) { a += sA[j]; b += sB[j]; c += sC[j]; }
    const float ia = rsqrtf(a), ib = rsqrtf(b);
    invA[row] = ia;
    invB[row] = ib;
    pos[row]  = c * ia * ib;
  }
}

// ---------------------------------------------------------------------------
// Kernel 2: TDM-staged fp32-WMMA GEMM + masked online log-sum-exp.
// Normalization applied in the epilogue (rank-1 scaling), so staging is a
// pure tile copy done by the Tensor Data Mover, double-buffered.
// ---------------------------------------------------------------------------
__global__ __launch_bounds__(256) void simlse_kernel(const float* __restrict__ A,
                                                     const float* __restrict__ B,
                                                     const float* __restrict__ invA,
                                                     const float* __restrict__ invB,
                                                     const float* __restrict__ pos,
                                                     float* __restrict__ per_row,
                                                     float* __restrict__ validf) {
  __shared__ float lA[2][TM * LDK];
  __shared__ float lB[2][TN * LDK];
  __shared__ float sM[8][TM], sS[8][TM], sN[8][TM];

  const int t       = threadIdx.x;
  const int lane    = t & 31;
  const int w       = t >> 5;     // wave 0..7 -> column sub-tile
  const int hf      = lane >> 4;  // lane half (K-pair / M+8 half)
  const int l16     = lane & 15;
  const int rowBase = blockIdx.x * TM;

  const unsigned ldsA0 = (unsigned)(uintptr_t)&lA[0][0];
  const unsigned ldsA1 = (unsigned)(uintptr_t)&lA[1][0];
  const unsigned ldsB0 = (unsigned)(uintptr_t)&lB[0][0];
  const unsigned ldsB1 = (unsigned)(uintptr_t)&lB[1][0];

  // per-row state, rows r = rowBase + mi*16 + hf*8 + v
  float mrow[2][8], srow[2][8], arow[2][8], prow[2][8], iAr[2][8];
#pragma unroll
  for (int mi = 0; mi < 2; ++mi)
#pragma unroll
    for (int v = 0; v < 8; ++v) {
      const int r = rowBase + mi * 16 + hf * 8 + v;
      mrow[mi][v] = NEG_INF;
      srow[mi][v] = 0.f;
      arow[mi][v] = 0.f;
      prow[mi][v] = pos[r];
      iAr[mi][v]  = invA[r];
    }

  const v8f vzero = {0.f, 0.f, 0.f, 0.f, 0.f, 0.f, 0.f, 0.f};

  for (int nt = 0; nt < BATCH; nt += TN) {
    v8f acc[2][2];
    acc[0][0] = vzero; acc[0][1] = vzero; acc[1][0] = vzero; acc[1][1] = vzero;

    // per-lane column inverse norms for the epilogue
    const float iB0 = invB[nt + w * 32 + l16];
    const float iB1 = invB[nt + w * 32 + 16 + l16];

    // prime the pipeline: DMA K-block 0 into buffer 0
    if (w == 0) {
      tdm_load_tile(A + (size_t)rowBase * DIM, ldsA0, TM);
      tdm_load_tile(B + (size_t)nt * DIM, ldsB0, TN);
    }

    for (int i = 0; i < NK; ++i) {
      if (w == 0) {
        if (i + 1 < NK) {  // prefetch next K-block into the other buffer
          const int knext = (i + 1) * KB;
          tdm_load_tile(A + (size_t)rowBase * DIM + knext,
                        (i + 1) & 1 ? ldsA1 : ldsA0, TM);
          tdm_load_tile(B + (size_t)nt * DIM + knext,
                        (i + 1) & 1 ? ldsB1 : ldsB0, TN);
          __builtin_amdgcn_s_wait_tensorcnt(2);  // current buffer complete
        } else {
          __builtin_amdgcn_s_wait_tensorcnt(0);  // drain
        }
      }
      __syncthreads();  // current buffer visible to all waves

      const float* cA = &lA[i & 1][0];
      const float* cB = &lB[i & 1][0];
#pragma unroll
      for (int kk = 0; kk < KB; kk += 4) {
        const int ko = kk + 2 * hf;  // A/B frag: K-pair base = 2*(lane>>4)
        v2f a0 = *(const v2f*)&cA[(l16)*LDK + ko];
        v2f a1 = *(const v2f*)&cA[(16 + l16) * LDK + ko];
        v2f b0 = *(const v2f*)&cB[(w * 32 + l16) * LDK + ko];
        v2f b1 = *(const v2f*)&cB[(w * 32 + 16 + l16) * LDK + ko];
        acc[0][0] = __builtin_amdgcn_wmma_f32_16x16x4_f32(false, a0, false, b0, (short)0, acc[0][0], false, false);
        acc[0][1] = __builtin_amdgcn_wmma_f32_16x16x4_f32(false, a0, false, b1, (short)0, acc[0][1], false, false);
        acc[1][0] = __builtin_amdgcn_wmma_f32_16x16x4_f32(false, a1, false, b0, (short)0, acc[1][0], false, false);
        acc[1][1] = __builtin_amdgcn_wmma_f32_16x16x4_f32(false, a1, false, b1, (short)0, acc[1][1], false, false);
      }
      __syncthreads();  // all reads done before TDM overwrites this buffer
    }

    // epilogue: normalize, hard-negative mask, online LSE (registers only)
#pragma unroll
    for (int mi = 0; mi < 2; ++mi)
#pragma unroll
      for (int ni = 0; ni < 2; ++ni)
#pragma unroll
        for (int v = 0; v < 8; ++v) {
          const float sim = acc[mi][ni][v] * iAr[mi][v] * (ni ? iB1 : iB0);
          const int r = rowBase + mi * 16 + hf * 8 + v;
          const int c = nt + w * 32 + ni * 16 + l16;
          const float p = prow[mi][v];
          const bool isdiag = (c == r);
          const bool inc = isdiag || ((sim <= p * TOPM) && (sim >= p * BOTM));
          const float z = isdiag ? p * INV_T : sim * INV_T;
          if (inc) {
            const float mo = mrow[mi][v];
            const float mn = fmaxf(mo, z);
            srow[mi][v] = srow[mi][v] * __expf(mo - mn) + __expf(z - mn);
            mrow[mi][v] = mn;
            arow[mi][v] += isdiag ? 0.f : 1.f;
          }
        }
  }

  // merge (m,s,cnt) across the 16 column-lanes (xor within 16-lane half)
#pragma unroll
  for (int mi = 0; mi < 2; ++mi)
#pragma unroll
    for (int v = 0; v < 8; ++v) {
      float m = mrow[mi][v], s = srow[mi][v], a = arow[mi][v];
#pragma unroll
      for (int msk = 1; msk <= 8; msk <<= 1) {
        const float mo = __shfl_xor(m, msk);
        const float so = __shfl_xor(s, msk);
        const float ao = __shfl_xor(a, msk);
        const float M  = fmaxf(m, mo);
        const float s1 = (m  > NEG_INF) ? s  * __expf(m  - M) : 0.f;
        const float s2 = (mo > NEG_INF) ? so * __expf(mo - M) : 0.f;
        m = M; s = s1 + s2; a += ao;
      }
      mrow[mi][v] = m; srow[mi][v] = s; arow[mi][v] = a;
    }

  if (l16 == 0) {
#pragma unroll
    for (int mi = 0; mi < 2; ++mi)
#pragma unroll
      for (int v = 0; v < 8; ++v) {
        const int rl = mi * 16 + hf * 8 + v;
        sM[w][rl] = mrow[mi][v];
        sS[w][rl] = srow[mi][v];
        sN[w][rl] = arow[mi][v];
      }
  }
  __syncthreads();

  if (t < TM) {  // merge across the 8 waves (each covered different columns)
    float M = NEG_INF, S = 0.f, Nn = 0.f;
#pragma unroll
    for (int wv = 0; wv < 8; ++wv) {
      const float m2 = sM[wv][t], s2 = sS[wv][t];
      const float Mx = fmaxf(M, m2);
      const float p1 = (M  > NEG_INF) ? S  * __expf(M  - Mx) : 0.f;
      const float p2 = (m2 > NEG_INF) ? s2 * __expf(m2 - Mx) : 0.f;
      M = Mx; S = p1 + p2; Nn += sN[wv][t];
    }
    const int r = rowBase + t;
    per_row[r] = M + __logf(S) - pos[r] * INV_T;
    validf[r]  = (Nn > 0.f) ? 1.f : 0.f;
  }
}

// ---------------------------------------------------------------------------
// Kernel 3: deterministic final reduction -> scalar loss
// ---------------------------------------------------------------------------
__global__ __launch_bounds__(256) void finalize_kernel(const float* __restrict__ per_row,
                                                       const float* __restrict__ validf,
                                                       float* __restrict__ out) {
  float tot = 0.f, cnt = 0.f;
  for (int i = threadIdx.x; i < BATCH; i += 256) {
    const float vl = validf[i];
    tot += vl * per_row[i];
    cnt += vl;
  }
#pragma unroll
  for (int off = 16; off > 0; off >>= 1) {
    tot += __shfl_xor(tot, off);
    cnt += __shfl_xor(cnt, off);
  }
  __shared__ float sT[8], sC[8];
  const int w = threadIdx.x >> 5, lane = threadIdx.x & 31;
  if (lane == 0) { sT[w] = tot; sC[w] = cnt; }
  __syncthreads();
  if (threadIdx.x == 0) {
    float T = 0.f, C = 0.f;
#pragma unroll
    for (int j = 0; j < 8; ++j) { T += sT[j]; C += sC[j]; }
    out[0] = T / fmaxf(C, 1.0f);
  }
}

// ---------------------------------------------------------------------------
extern "C" void kernel_launch(void* const* d_in, const int* in_sizes, int n_in,
                              void* d_out, int out_size, void* d_ws, size_t ws_size,
                              hipStream_t stream) {
  const float* A = (const float*)d_in[0];
  const float* B = (const float*)d_in[1];
  float* ws      = (float*)d_ws;
  float* invA    = ws;
  float* invB    = ws + BATCH;
  float* pos     = ws + 2 * BATCH;
  float* per_row = ws + 3 * BATCH;
  float* validf  = ws + 4 * BATCH;

  norms_kernel<<<BATCH, 256, 0, stream>>>(A, B, invA, invB, pos);
  simlse_kernel<<<BATCH / TM, 256, 0, stream>>>(A, B, invA, invB, pos, per_row, validf);
  finalize_kernel<<<1, 256, 0, stream>>>(per_row, validf, (float*)d_out);
}